// IMP_GCN_21303037788345
// MI455X (gfx1250) — compile-verified
//
#include <hip/hip_runtime.h>

// Problem constants (match reference)
constexpr int NU  = 100000;
constexpr int NA  = 25000;
constexpr int NN  = NU + NA;      // 125000
constexpr int DD  = 128;
constexpr int GG  = 4;
constexpr long NNZ = 2000000;
constexpr float LAYER_W = 0.2f;

typedef __attribute__((ext_vector_type(2))) float v2f;
typedef __attribute__((ext_vector_type(8))) float v8f;

// ---------------------------------------------------------------------------
// dst[i] = scale * ego[i]   (ego = concat(emb_user, emb_uploader), flat)
// ---------------------------------------------------------------------------
__global__ void __launch_bounds__(256)
k_init_ego(const float* __restrict__ eu, const float* __restrict__ ea,
           float* __restrict__ dst, float scale) {
  long i = (long)blockIdx.x * 256 + threadIdx.x;
  if (i >= (long)NN * DD) return;
  float v = (i < (long)NU * DD) ? eu[i] : ea[i - (long)NU * DD];
  dst[i] = scale * v;
}

__global__ void __launch_bounds__(256)
k_zero(float* __restrict__ dst) {
  long i = (long)blockIdx.x * 256 + threadIdx.x;
  if (i < (long)NN * DD) dst[i] = 0.0f;
}

__global__ void __launch_bounds__(256)
k_accum(float* __restrict__ out, const float* __restrict__ x) {
  long i = (long)blockIdx.x * 256 + threadIdx.x;
  if (i < (long)NN * DD) out[i] += LAYER_W * x[i];
}

// ---------------------------------------------------------------------------
// side-spmm fused into h:  h[row[e]] += vals[e] * ego[col[e]]
// 32 lanes per edge, 4 floats per lane (float4 gather, 4 f32 atomics)
// ---------------------------------------------------------------------------
__global__ void __launch_bounds__(256)
k_scatter_ego(const float* __restrict__ vals, const int* __restrict__ row,
              const int* __restrict__ col, const float* __restrict__ eu,
              const float* __restrict__ ea, float* __restrict__ y) {
  long gid = (long)blockIdx.x * 256 + threadIdx.x;
  long e = gid >> 5;
  int lane = (int)(gid & 31);
  if (e >= NNZ) return;
  int r = row[e], c = col[e];
  float v = vals[e];
  const float* src = (c < NU) ? (eu + (long)c * DD) : (ea + (long)(c - NU) * DD);
  float4 x = *(const float4*)(src + lane * 4);
  float* dst = y + (long)r * DD + lane * 4;
  atomicAdd(dst + 0, v * x.x);
  atomicAdd(dst + 1, v * x.y);
  atomicAdd(dst + 2, v * x.z);
  atomicAdd(dst + 3, v * x.w);
}

// masked propagation layer: xnext[row[e]] += vals[e]*m[row]*m[col] * xcur[col[e]]
__global__ void __launch_bounds__(256)
k_scatter_masked(const float* __restrict__ vals, const int* __restrict__ row,
                 const int* __restrict__ col, const float* __restrict__ mask,
                 int g, const float* __restrict__ xcur, float* __restrict__ xnext) {
  long gid = (long)blockIdx.x * 256 + threadIdx.x;
  long e = gid >> 5;
  int lane = (int)(gid & 31);
  if (e >= NNZ) return;
  int r = row[e], c = col[e];
  float v = vals[e] * mask[(long)r * GG + g] * mask[(long)c * GG + g];
  if (v == 0.0f) return;  // wave-uniform per edge (32 lanes == one edge)
  float4 x = *(const float4*)(xcur + (long)c * DD + lane * 4);
  float* dst = xnext + (long)r * DD + lane * 4;
  atomicAdd(dst + 0, v * x.x);
  atomicAdd(dst + 1, v * x.y);
  atomicAdd(dst + 2, v * x.z);
  atomicAdd(dst + 3, v * x.w);
}

// ---------------------------------------------------------------------------
// temp = leaky_relu(h @ W + b, 0.01)   via V_WMMA_F32_16X16X4_F32
// Block: 256 threads = 8 wave32; each wave owns a 16-row x 128-col strip.
// W staged in LDS packed as K-pairs:
//   sWp[(k>>1)*256 + n*2 + (k&1)] = W[k][n]
// so a B fragment {W[ka][n], W[ka+1][n]} is one aligned ds_load_b64.
// A frag (16x4 f32): lanes 0-15 hold K={k,k+1}, lanes 16-31 K={k+2,k+3};
// row address is clamped (not predicated): garbage A rows only affect
// D rows >= N, which are discarded at the store.
// C/D layout: col = lane%16, row = vgpr + 8*(lane>=16).
// ---------------------------------------------------------------------------
__global__ void __launch_bounds__(256)
k_fc_wmma(const float* __restrict__ h, const float* __restrict__ W,
          const float* __restrict__ bias, float* __restrict__ temp) {
  __shared__ float sWp[DD * DD];  // 64 KB, K-pair packed
  for (int i4 = threadIdx.x; i4 < DD * DD / 4; i4 += 256) {
    const int k = i4 >> 5;          // source row of W
    const int n = (i4 & 31) * 4;    // source col (x4)
    float4 w = ((const float4*)W)[i4];
    float* d = sWp + (k >> 1) * 256 + (k & 1);
    d[(n + 0) * 2] = w.x;
    d[(n + 1) * 2] = w.y;
    d[(n + 2) * 2] = w.z;
    d[(n + 3) * 2] = w.w;
  }
  __syncthreads();

  const int wave = threadIdx.x >> 5;
  const int lane = threadIdx.x & 31;
  const int half = lane >> 4;
  const int l16  = lane & 15;
  const long base = (long)blockIdx.x * 128 + wave * 16;  // first row of strip
  const long rowA = base + l16;
  const long rowLd = (rowA < NN) ? rowA : (long)(NN - 1);  // clamp, no predication
  const float* __restrict__ hrow = h + rowLd * DD;

  v8f acc[8] = {};
  for (int k = 0; k < DD; k += 4) {
    const int ka = k + 2 * half;                       // even
    v2f a = *(const v2f*)(hrow + ka);                  // global_load_b64
    const float* sWrow = sWp + (ka >> 1) * 256;
#pragma unroll
    for (int t = 0; t < 8; ++t) {
      v2f bb = *(const v2f*)(sWrow + (t * 16 + l16) * 2);  // ds_load_b64
      acc[t] = __builtin_amdgcn_wmma_f32_16x16x4_f32(
          false, a, false, bb, (short)0, acc[t], false, false);
    }
  }

#pragma unroll
  for (int t = 0; t < 8; ++t) {
    const int colN = t * 16 + l16;
    const float bc = bias[colN];
#pragma unroll
    for (int v = 0; v < 8; ++v) {
      const long m = base + half * 8 + v;
      if (m < NN) {
        float x = acc[t][v] + bc;
        x = (x > 0.0f) ? x : 0.01f * x;
        temp[m * DD + colN] = x;
      }
    }
  }
}

// ---------------------------------------------------------------------------
// scores = temp @ fcg_W + fcg_b ; mask = tie-aware top-1 one-hot (users),
// all-ones (uploaders). One thread per row.
// ---------------------------------------------------------------------------
__global__ void __launch_bounds__(256)
k_scores(const float* __restrict__ temp, const float* __restrict__ Wg,
         const float* __restrict__ bg, float* __restrict__ mask) {
  long r = (long)blockIdx.x * 256 + threadIdx.x;
  if (r >= NN) return;
  if (r >= NU) {
    mask[r * GG + 0] = 1.0f; mask[r * GG + 1] = 1.0f;
    mask[r * GG + 2] = 1.0f; mask[r * GG + 3] = 1.0f;
    return;
  }
  float s0 = bg[0], s1 = bg[1], s2 = bg[2], s3 = bg[3];
  const float* tr = temp + r * DD;
  for (int d = 0; d < DD; ++d) {
    float tv = tr[d];
    float4 w = ((const float4*)Wg)[d];  // Wg is [D,4] row-major
    s0 += tv * w.x; s1 += tv * w.y; s2 += tv * w.z; s3 += tv * w.w;
  }
  float mx = fmaxf(fmaxf(s0, s1), fmaxf(s2, s3));
  mask[r * GG + 0] = (s0 == mx) ? 1.0f : 0.0f;
  mask[r * GG + 1] = (s1 == mx) ? 1.0f : 0.0f;
  mask[r * GG + 2] = (s2 == mx) ? 1.0f : 0.0f;
  mask[r * GG + 3] = (s3 == mx) ? 1.0f : 0.0f;
}

// ---------------------------------------------------------------------------
extern "C" void kernel_launch(void* const* d_in, const int* in_sizes, int n_in,
                              void* d_out, int out_size, void* d_ws, size_t ws_size,
                              hipStream_t stream) {
  const float* eu   = (const float*)d_in[0];  // emb_user     [NU,D]
  const float* ea   = (const float*)d_in[1];  // emb_uploader [NA,D]
  const float* fcW  = (const float*)d_in[2];  // [D,D]
  const float* fcb  = (const float*)d_in[3];  // [D]
  const float* fcgW = (const float*)d_in[4];  // [D,G]
  const float* fcgb = (const float*)d_in[5];  // [G]
  const float* ev   = (const float*)d_in[6];  // [NNZ]
  const int*   er   = (const int*)d_in[7];    // [NNZ]
  const int*   ec   = (const int*)d_in[8];    // [NNZ]
  float* out = (float*)d_out;                 // [N,D] flat

  const size_t NDsz = (size_t)NN * DD;        // 16,000,000 elems
  float* wsA  = (float*)d_ws;                 // h, then x ping
  float* wsB  = wsA + NDsz;                   // temp, then x pong
  float* mask = wsB + NDsz;                   // [N,G]

  const int gND   = (int)((NDsz + 255) / 256);          // 62,500
  const int gEdge = (int)((NNZ * 32 + 255) / 256);      // 250,000
  const int gRow  = (NN + 255) / 256;
  const int gGemm = (NN + 127) / 128;

  // h = ego + side
  k_init_ego<<<gND, 256, 0, stream>>>(eu, ea, wsA, 1.0f);
  k_scatter_ego<<<gEdge, 256, 0, stream>>>(ev, er, ec, eu, ea, wsA);
  // temp = leaky_relu(h @ fc_W + fc_b) ; mask from scores
  k_fc_wmma<<<gGemm, 256, 0, stream>>>(wsA, fcW, fcb, wsB);
  k_scores<<<gRow, 256, 0, stream>>>(wsB, fcgW, fcgb, mask);

  // out = 0.2 * G * ego
  k_init_ego<<<gND, 256, 0, stream>>>(eu, ea, out, LAYER_W * GG);

  // 4 propagation layers per group, groups sequential (ws = 2 buffers)
  for (int g = 0; g < GG; ++g) {
    float* xA = wsA;
    float* xB = wsB;
    k_init_ego<<<gND, 256, 0, stream>>>(eu, ea, xA, 1.0f);
    for (int l = 1; l < 5; ++l) {
      k_zero<<<gND, 256, 0, stream>>>(xB);
      k_scatter_masked<<<gEdge, 256, 0, stream>>>(ev, er, ec, mask, g, xA, xB);
      k_accum<<<gND, 256, 0, stream>>>(out, xB);
      float* t = xA; xA = xB; xB = t;
    }
  }
}